// ForecastTKGQA_MHS_17403207483556
// MI455X (gfx1250) — compile-verified
//
#include <hip/hip_runtime.h>
#include <hip/hip_bf16.h>
#include <stdint.h>

// Problem constants (from reference)
#define Bc    128
#define Pc    512
#define Dc    200
#define RANKc 100
#define Ec    20575
#define BE    (Bc * Ec)          // 2,633,600

// Scoring kernel decomposition
#define GROUPS         8         // p-groups per batch row
#define WAVES          8         // waves per block (256 threads, wave32)
#define PAIRS_PER_WAVE (Pc / (GROUPS * WAVES))   // 8
#define Q4             (RANKc / 4)               // 25 float4 per half-row

typedef unsigned int v4u __attribute__((ext_vector_type(4)));
typedef int          v8i __attribute__((ext_vector_type(8)));
typedef int          v4i __attribute__((ext_vector_type(4)));

// ---------------------------------------------------------------------------
// Kernel 1: zero the workspace (sums1, cnt1, sums2, cnt2 = 4*BE floats)
// ---------------------------------------------------------------------------
__global__ void tkgqa_zero_ws(float4* __restrict__ ws, int n4) {
    int i = blockIdx.x * blockDim.x + threadIdx.x;
    if (i < n4) {
        float4 z; z.x = 0.f; z.y = 0.f; z.z = 0.f; z.w = 0.f;
        ws[i] = z;
    }
}

// ---------------------------------------------------------------------------
// Per-component score accumulation:
//   A  = l0*r0 - l1*r1 ; Bv = l1*r0 + l0*r1
//   acc += A*(q0*h0 + q1*h1) + Bv*(q0*h1 - q1*h0)
// ---------------------------------------------------------------------------
__device__ __forceinline__ void accum4(const float4 l0, const float4 l1,
                                       const float4 r0, const float4 r1,
                                       const float4 h0, const float4 h1,
                                       const float4 q0, const float4 q1,
                                       float& acc) {
#define ACC_C(c)                                                    \
    {                                                               \
        float A  = fmaf(l0.c, r0.c, -(l1.c * r1.c));                \
        float Bv = fmaf(l1.c, r0.c,  (l0.c * r1.c));                \
        float c0 = fmaf(q0.c, h0.c,  (q1.c * h1.c));                \
        float c1 = fmaf(q0.c, h1.c, -(q1.c * h0.c));                \
        acc = fmaf(A, c0, fmaf(Bv, c1, acc));                       \
    }
    ACC_C(x) ACC_C(y) ACC_C(z) ACC_C(w)
#undef ACC_C
}

// Prefetch one 800B embedding row: 32 lanes * 32B stride covers the row.
__device__ __forceinline__ void prefetch_row(const float* __restrict__ tkg,
                                             int row, int lane) {
    const char* p = (const char*)(tkg + (size_t)row * Dc);
    __builtin_prefetch(p + lane * 32, 0, 3);   // -> global_prefetch_b8
}

// ---------------------------------------------------------------------------
// Kernel 2: compute both scores per (b,p), scatter-add sums & counts.
// One block per (b, p-group). q[b] row staged into LDS via the Tensor Data
// Mover (async DMA, TENSORcnt). Next pair's rows are prefetched while the
// current pair computes (software pipeline over the gather latency).
// ---------------------------------------------------------------------------
__global__ __launch_bounds__(256)
void tkgqa_score(const float* __restrict__ tkg,
                 const float* __restrict__ qemb,
                 const int* __restrict__ s_idx,  const int* __restrict__ r1_idx,
                 const int* __restrict__ n1_idx, const int* __restrict__ r2_idx,
                 const int* __restrict__ n2_idx,
                 const int* __restrict__ tail1,  const int* __restrict__ tail2,
                 float* __restrict__ sums1, float* __restrict__ cnt1,
                 float* __restrict__ sums2, float* __restrict__ cnt2) {
    __shared__ float sh_q[Dc];

    const int b  = blockIdx.x / GROUPS;
    const int pg = blockIdx.x % GROUPS;

#if defined(__has_builtin) && __has_builtin(__builtin_amdgcn_tensor_load_to_lds)
    // --- TDM: DMA q row (200 f32 = 800B) from global into LDS ---
    if (threadIdx.x < 32) {
        const float* qrow = qemb + (size_t)b * Dc;
        unsigned long long gaddr = (unsigned long long)(uintptr_t)qrow;
        unsigned int lds_off = (unsigned int)(uintptr_t)(void*)&sh_q[0]; // low 32b of flat = LDS offset

        // D# group0: count=1, lds_addr, global_addr[56:0], type=2
        v4u g0 = { 1u,
                   lds_off,
                   (unsigned int)gaddr,
                   (((unsigned int)(gaddr >> 32)) & 0x01FFFFFFu) | (2u << 30) };
        // D# group1: data_size=4B, tensor_dim0=200, tensor_dim1=1,
        //            tile_dim0=200, tile_dim1=1, dim0_stride=200
        v8i g1 = { (int)(2u << 16),        // workgroup_mask=0, data_size=2 (4B)
                   (int)(200u << 16),      // tensor_dim0[15:0]
                   (int)(1u << 16),        // tensor_dim0 hi=0, tensor_dim1=1
                   (int)(200u << 16),      // tensor_dim1 hi=0, tile_dim0=200
                   1,                      // tile_dim1=1, tile_dim2=0
                   200,                    // tensor_dim0_stride[31:0]
                   0, 0 };
        v4i z4 = { 0, 0, 0, 0 };
#if __clang_major__ >= 23
        v8i z8 = { 0, 0, 0, 0, 0, 0, 0, 0 };
        __builtin_amdgcn_tensor_load_to_lds(g0, g1, z4, z4, z8, 0);
#else
        __builtin_amdgcn_tensor_load_to_lds(g0, g1, z4, z4, 0);
#endif
        __builtin_amdgcn_s_wait_tensorcnt(0);
    }
#else
    if (threadIdx.x < Dc) sh_q[threadIdx.x] = qemb[(size_t)b * Dc + threadIdx.x];
#endif
    __syncthreads();

    const int wave = threadIdx.x >> 5;
    const int lane = threadIdx.x & 31;
    const bool active = lane < Q4;   // 25 lanes * float4 = 100 ranks

    const float4* sq = (const float4*)sh_q;
    float4 q0{}, q1{};
    if (active) { q0 = sq[lane]; q1 = sq[Q4 + lane]; }

    const int pbase = pg * (WAVES * PAIRS_PER_WAVE) + wave * PAIRS_PER_WAVE;
    const int bp0   = b * Pc + pbase;

    // Pipeline prologue: indices + prefetch for the first pair.
    int si  = s_idx[bp0];
    int r1i = r1_idx[bp0];
    int n1i = n1_idx[bp0];
    int r2i = r2_idx[bp0];
    int n2i = n2_idx[bp0];
    prefetch_row(tkg, si,  lane);
    prefetch_row(tkg, r1i, lane);
    prefetch_row(tkg, n1i, lane);
    prefetch_row(tkg, r2i, lane);
    prefetch_row(tkg, n2i, lane);

#pragma unroll 1
    for (int j = 0; j < PAIRS_PER_WAVE; ++j) {
        const int bp = bp0 + j;

        // Prefetch next pair's rows while this pair's gathers are in flight.
        int nsi = 0, nr1 = 0, nn1 = 0, nr2 = 0, nn2 = 0;
        if (j + 1 < PAIRS_PER_WAVE) {
            const int nbp = bp + 1;
            nsi = s_idx[nbp];  nr1 = r1_idx[nbp]; nn1 = n1_idx[nbp];
            nr2 = r2_idx[nbp]; nn2 = n2_idx[nbp];
            prefetch_row(tkg, nsi, lane);
            prefetch_row(tkg, nr1, lane);
            prefetch_row(tkg, nn1, lane);
            prefetch_row(tkg, nr2, lane);
            prefetch_row(tkg, nn2, lane);
        }

        float acc1 = 0.f, acc2 = 0.f;
        if (active) {
            const float4* S  = (const float4*)(tkg + (size_t)si  * Dc);
            const float4* R1 = (const float4*)(tkg + (size_t)r1i * Dc);
            const float4* N1 = (const float4*)(tkg + (size_t)n1i * Dc);
            const float4* R2 = (const float4*)(tkg + (size_t)r2i * Dc);
            const float4* N2 = (const float4*)(tkg + (size_t)n2i * Dc);

            float4 s0 = S[lane],   s1 = S[Q4 + lane];
            float4 a0 = R1[lane],  a1 = R1[Q4 + lane];
            float4 h0 = N1[lane],  h1 = N1[Q4 + lane];   // shared between scores
            float4 c0 = R2[lane],  c1 = R2[Q4 + lane];
            float4 g0v = N2[lane], g1v = N2[Q4 + lane];

            accum4(s0, s1, a0, a1, h0, h1, q0, q1, acc1);   // score_one(s, r1, n1, q)
            accum4(h0, h1, c0, c1, g0v, g1v, q0, q1, acc2); // score_two(n1, r2, n2, q)
        }

        // wave32 butterfly reduction (lowers to ds_swizzle_b32)
#pragma unroll
        for (int off = 16; off > 0; off >>= 1) {
            acc1 += __shfl_xor(acc1, off, 32);
            acc2 += __shfl_xor(acc2, off, 32);
        }

        if (lane == 0) {
            const int t1 = tail1[bp];
            const int t2 = tail2[bp];
            atomicAdd(&sums1[(size_t)b * Ec + t1], acc1);
            atomicAdd(&cnt1 [(size_t)b * Ec + t1], 1.0f);
            atomicAdd(&sums2[(size_t)b * Ec + t2], acc2);
            atomicAdd(&cnt2 [(size_t)b * Ec + t2], 1.0f);
        }

        // Rotate pipeline registers.
        si = nsi; r1i = nr1; n1i = nn1; r2i = nr2; n2i = nn2;
    }
}

// ---------------------------------------------------------------------------
// Kernel 3: finalize   out = 0.8*(0.8 + mean1) + mean2 ;  (==0) -> -99999
// Vectorized float4 (BE % 4 == 0): 52.6 MB streamed, HBM-bound.
// ---------------------------------------------------------------------------
__global__ void tkgqa_finalize(const float4* __restrict__ sums1,
                               const float4* __restrict__ cnt1,
                               const float4* __restrict__ sums2,
                               const float4* __restrict__ cnt2,
                               float4* __restrict__ out, int n4) {
    int i = blockIdx.x * blockDim.x + threadIdx.x;
    if (i < n4) {
        float4 s1 = sums1[i], c1 = cnt1[i];
        float4 s2 = sums2[i], c2 = cnt2[i];
        float4 o;
#define FIN_C(c)                                                          \
        {                                                                 \
            float m1 = (c1.c > 0.f) ? (s1.c / fmaxf(c1.c, 1.f)) : 0.f;    \
            float m2 = (c2.c > 0.f) ? (s2.c / fmaxf(c2.c, 1.f)) : 0.f;    \
            float v = 0.8f * (0.8f + m1) + m2;                            \
            o.c = (v == 0.0f) ? -99999.0f : v;                            \
        }
        FIN_C(x) FIN_C(y) FIN_C(z) FIN_C(w)
#undef FIN_C
        out[i] = o;
    }
}

// ---------------------------------------------------------------------------
extern "C" void kernel_launch(void* const* d_in, const int* in_sizes, int n_in,
                              void* d_out, int out_size, void* d_ws, size_t ws_size,
                              hipStream_t stream) {
    const float* tkg  = (const float*)d_in[0];
    const float* qemb = (const float*)d_in[1];
    const int* s_idx  = (const int*)d_in[2];
    const int* r1_idx = (const int*)d_in[3];
    const int* n1_idx = (const int*)d_in[4];
    const int* r2_idx = (const int*)d_in[5];
    const int* n2_idx = (const int*)d_in[6];
    const int* tail1  = (const int*)d_in[7];
    const int* tail2  = (const int*)d_in[8];
    float* out = (float*)d_out;

    // workspace layout: sums1 | cnt1 | sums2 | cnt2 (each BE floats, ~42 MB)
    float* sums1 = (float*)d_ws;
    float* cnt1  = sums1 + BE;
    float* sums2 = cnt1 + BE;
    float* cnt2  = sums2 + BE;

    const int nz4 = 4 * BE / 4;                 // float4 count of whole ws
    tkgqa_zero_ws<<<(nz4 + 255) / 256, 256, 0, stream>>>((float4*)d_ws, nz4);

    tkgqa_score<<<Bc * GROUPS, 256, 0, stream>>>(
        tkg, qemb, s_idx, r1_idx, n1_idx, r2_idx, n2_idx, tail1, tail2,
        sums1, cnt1, sums2, cnt2);

    const int nf4 = BE / 4;                     // 658,400
    tkgqa_finalize<<<(nf4 + 255) / 256, 256, 0, stream>>>(
        (const float4*)sums1, (const float4*)cnt1,
        (const float4*)sums2, (const float4*)cnt2, (float4*)out, nf4);
}